// mtclf_39822936769202
// MI455X (gfx1250) — compile-verified
//
#include <hip/hip_runtime.h>
#include <hip/hip_bf16.h>

// ---------------------------------------------------------------------------
// Multi-task GraphSAGE stack for MI455X (gfx1250, wave32).
// GEMMs run on V_WMMA_F32_16X16X4_F32; full-width layers use an unguarded
// 16x32-per-wave slab (2 accumulators share one A load), padded heads use a
// guarded 16x16 path. Edge aggregation via f32 global atomics.
// ---------------------------------------------------------------------------

#define N_NODES 50000
#define N_EDGES 800000

typedef float v2f __attribute__((ext_vector_type(2)));
typedef float v8f __attribute__((ext_vector_type(8)));

// ---------------- degree / inverse degree ----------------------------------
__global__ void deg_kernel(const int* __restrict__ dst, float* __restrict__ deg,
                           int n_edges) {
    int e = blockIdx.x * blockDim.x + threadIdx.x;
    if (e < n_edges) atomicAdd(&deg[dst[e]], 1.0f);
}

__global__ void invdeg_kernel(float* __restrict__ deg, int n) {
    int i = blockIdx.x * blockDim.x + threadIdx.x;
    if (i < n) deg[i] = 1.0f / fmaxf(deg[i], 1.0f);
}

// ---------------- edge scatter-add (segment_sum of x[src] into agg[dst]) ---
__global__ void scatter_kernel(const float* __restrict__ x,
                               const int* __restrict__ src,
                               const int* __restrict__ dst,
                               float* __restrict__ agg,
                               int n_edges, int Fvec /*F/4*/, int F) {
    int idx = blockIdx.x * blockDim.x + threadIdx.x;
    int total = n_edges * Fvec;
    if (idx >= total) return;
    int e  = idx / Fvec;
    int f4 = (idx - e * Fvec) * 4;
    int s = src[e], d = dst[e];
    const float4 v = *(const float4*)(x + (size_t)s * F + f4);
    float* a = agg + (size_t)d * F + f4;
    atomicAdd(a + 0, v.x);
    atomicAdd(a + 1, v.y);
    atomicAdd(a + 2, v.z);
    atomicAdd(a + 3, v.w);
}

// ---------------- fused SAGE GEMM:  out = (agg*inv_deg)@Wl + bl + x@Wr -----
// ACT: 0 = none, 1 = relu, 2 = sigmoid
// NT:  16-column tiles per wave (1 or 2). GUARD: pad columns beyond Nout.
template <int ACT, int NT, bool GUARD>
__global__ void sage_gemm(const float* __restrict__ agg,    // [n, K]
                          const float* __restrict__ xr,     // [n, K]
                          const float* __restrict__ invdeg, // [n]
                          const float* __restrict__ Wl,     // [K, Nout]
                          const float* __restrict__ bl,     // [Nout]
                          const float* __restrict__ Wr,     // [K, Nout]
                          float* __restrict__ out,          // [n, outStride]
                          int outStride, int n, int K, int Nout) {
    const int lane  = threadIdx.x & 31;
    const int half  = lane >> 4;   // selects K pair within 4-wide step
    const int l16   = lane & 15;
    const int wavesPerBlock = blockDim.x >> 5;
    const int wave  = blockIdx.x * wavesPerBlock + (threadIdx.x >> 5);
    const int totalWaves = gridDim.x * wavesPerBlock;

    const int tilesM = n >> 4;                   // n multiple of 16 (50000)
    const int tilesN = (Nout + 15) >> 4;
    const int nSlabs = tilesN / NT;              // NT divides tilesN (dispatch)
    const int nTiles = tilesM * nSlabs;

    for (int t = wave; t < nTiles; t += totalWaves) {
        const int tm = t / nSlabs;
        const int sn = t - tm * nSlabs;

        const int row = (tm << 4) + l16;         // A row for this lane
        const float id = invdeg[row];
        const float* __restrict__ aggRow = agg + (size_t)row * K;
        const float* __restrict__ xRow   = xr  + (size_t)row * K;

        int  col[NT], colc[NT];
        bool ok[NT];
#pragma unroll
        for (int j = 0; j < NT; ++j) {
            col[j]  = ((sn * NT + j) << 4) + l16;
            ok[j]   = GUARD ? (col[j] < Nout) : true;
            colc[j] = (GUARD && !ok[j]) ? (Nout - 1) : col[j];
        }

        // prefetch next slab's A rows (global_prefetch_b8 path)
        const int tNext = t + totalWaves;
        if (tNext < nTiles) {
            const int tmN = tNext / nSlabs;
            __builtin_prefetch(agg + (size_t)((tmN << 4) + l16) * K, 0, 0);
            __builtin_prefetch(xr  + (size_t)((tmN << 4) + l16) * K, 0, 0);
        }

        v8f c[NT];
#pragma unroll
        for (int j = 0; j < NT; ++j) c[j] = (v8f){};

        for (int k0 = 0; k0 < K; k0 += 4) {
            const int ka = k0 + (half << 1);     // this lane's K pair (even)
            const float2 va = *(const float2*)(aggRow + ka);
            const float2 vx = *(const float2*)(xRow + ka);
            v2f aA, aX;
            aA[0] = va.x * id;  aA[1] = va.y * id;
            aX[0] = vx.x;       aX[1] = vx.y;
#pragma unroll
            for (int j = 0; j < NT; ++j) {
                const float* wl = Wl + (size_t)ka * Nout + colc[j];
                const float* wr = Wr + (size_t)ka * Nout + colc[j];
                v2f bL, bR;
                bL[0] = wl[0]; bL[1] = wl[Nout];
                bR[0] = wr[0]; bR[1] = wr[Nout];
                if (GUARD && !ok[j]) { bL[0] = 0.f; bL[1] = 0.f;
                                       bR[0] = 0.f; bR[1] = 0.f; }
                c[j] = __builtin_amdgcn_wmma_f32_16x16x4_f32(
                           false, aA, false, bL, (short)0, c[j], false, false);
                c[j] = __builtin_amdgcn_wmma_f32_16x16x4_f32(
                           false, aX, false, bR, (short)0, c[j], false, false);
            }
        }

#pragma unroll
        for (int j = 0; j < NT; ++j) {
            const float bias = bl[colc[j]];
#pragma unroll
            for (int i = 0; i < 8; ++i) {
                const int orow = (tm << 4) + (half << 3) + i;   // C/D layout
                float v = c[j][i] + bias;
                if (ACT == 1) v = fmaxf(v, 0.0f);
                else if (ACT == 2) v = 1.0f / (1.0f + __expf(-v));
                if (!GUARD || ok[j])
                    out[(size_t)orow * outStride + col[j]] = v;
            }
        }
    }
}

// ---------------- cross-entropy on sigmoid logits --------------------------
__global__ void ce_kernel(const float* __restrict__ o, int oStride,
                          const int* __restrict__ ylab, // pre-offset, stride 3
                          int C, int n, float* __restrict__ out) {
    int i = blockIdx.x * blockDim.x + threadIdx.x;
    float contrib = 0.0f;
    if (i < n) {
        const float* l = o + (size_t)i * oStride;
        int lab = ylab[(size_t)i * 3];
        float m = -1e30f;
        for (int c = 0; c < C; ++c) m = fmaxf(m, l[c]);
        float s = 0.0f;
        for (int c = 0; c < C; ++c) s += __expf(l[c] - m);
        float lse = m + __logf(s);
        contrib = -(l[lab] - lse) / (float)n;
    }
    // wave32 reduction, one atomic per wave
    for (int off = 16; off > 0; off >>= 1)
        contrib += __shfl_down(contrib, off, 32);
    if ((threadIdx.x & 31) == 0) atomicAdd(out, contrib);
}

// ---------------------------------------------------------------------------
// Host-side helpers
// ---------------------------------------------------------------------------
static void launch_gemm(int act, const float* agg, const float* x,
                        const float* invdeg, const float* Wl, const float* bl,
                        const float* Wr, float* out, int outStride,
                        int K, int Nout, hipStream_t stream) {
    if ((Nout & 31) == 0) {
        // full-width layers: 16x32 slab per wave, no column guard
        int nTiles = (N_NODES / 16) * (Nout / 32);
        int blocks = (nTiles + 7) / 8;   // 8 waves (256 threads) per block
        if (act == 1)
            sage_gemm<1, 2, false><<<blocks, 256, 0, stream>>>(
                agg, x, invdeg, Wl, bl, Wr, out, outStride, N_NODES, K, Nout);
        else
            sage_gemm<2, 2, false><<<blocks, 256, 0, stream>>>(
                agg, x, invdeg, Wl, bl, Wr, out, outStride, N_NODES, K, Nout);
    } else {
        // padded heads: guarded 16x16 tile per wave
        int nTiles = (N_NODES / 16) * ((Nout + 15) / 16);
        int blocks = (nTiles + 7) / 8;
        sage_gemm<2, 1, true><<<blocks, 256, 0, stream>>>(
            agg, x, invdeg, Wl, bl, Wr, out, outStride, N_NODES, K, Nout);
    }
}

static void launch_agg(const float* x, int F, const int* esrc, const int* edst,
                       float* aggbuf, hipStream_t stream) {
    hipMemsetAsync(aggbuf, 0, (size_t)N_NODES * F * sizeof(float), stream);
    int total = N_EDGES * (F / 4);
    scatter_kernel<<<(total + 255) / 256, 256, 0, stream>>>(
        x, esrc, edst, aggbuf, N_EDGES, F / 4, F);
}

extern "C" void kernel_launch(void* const* d_in, const int* in_sizes, int n_in,
                              void* d_out, int out_size, void* d_ws, size_t ws_size,
                              hipStream_t stream) {
    const float* X     = (const float*)d_in[0];
    const int*   esrc  = (const int*)d_in[1];
    const int*   edst  = (const int*)d_in[2];
    const int*   y     = (const int*)d_in[3];
    const float* Wl_i0 = (const float*)d_in[4];
    const float* bl_i0 = (const float*)d_in[5];
    const float* Wr_i0 = (const float*)d_in[6];
    const float* Wl_i1 = (const float*)d_in[7];
    const float* bl_i1 = (const float*)d_in[8];
    const float* Wr_i1 = (const float*)d_in[9];
    const float* Wl_m  = (const float*)d_in[10];
    const float* bl_m  = (const float*)d_in[11];
    const float* Wr_m  = (const float*)d_in[12];
    const float* Wl_o  = (const float*)d_in[13];
    const float* bl_o  = (const float*)d_in[14];
    const float* Wr_o  = (const float*)d_in[15];
    const float* Wl_h[3] = {(const float*)d_in[16], (const float*)d_in[19], (const float*)d_in[22]};
    const float* bl_h[3] = {(const float*)d_in[17], (const float*)d_in[20], (const float*)d_in[23]};
    const float* Wr_h[3] = {(const float*)d_in[18], (const float*)d_in[21], (const float*)d_in[24]};
    const int    C_h[3]  = {12, 8, 5};

    // Workspace layout (floats): invdeg | agg[N*256] | xk[N*256] | xk2[N*256]
    float* ws     = (float*)d_ws;
    float* invdeg = ws;
    float* agg    = ws + 50048;                 // keep 16B alignment
    float* xk     = agg + (size_t)N_NODES * 256;
    float* xk2    = xk  + (size_t)N_NODES * 256;

    // degrees (computed once, shared across tasks)
    hipMemsetAsync(invdeg, 0, N_NODES * sizeof(float), stream);
    deg_kernel<<<(N_EDGES + 255) / 256, 256, 0, stream>>>(edst, invdeg, N_EDGES);
    invdeg_kernel<<<(N_NODES + 255) / 256, 256, 0, stream>>>(invdeg, N_NODES);

    hipMemsetAsync(d_out, 0, sizeof(float), stream);

    for (int k = 0; k < 2; ++k) {
        const float* x0 = X + ((size_t)(2 * k + 0)) * N_NODES * 128;
        const float* x1 = X + ((size_t)(2 * k + 1)) * N_NODES * 128;

        // input SAGE layers -> xk = concat(h0, h1) [N, 256]
        launch_agg(x0, 128, esrc, edst, agg, stream);
        launch_gemm(1, agg, x0, invdeg, Wl_i0, bl_i0, Wr_i0, xk, 256, 128, 128, stream);
        launch_agg(x1, 128, esrc, edst, agg, stream);
        launch_gemm(1, agg, x1, invdeg, Wl_i1, bl_i1, Wr_i1, xk + 128, 256, 128, 128, stream);

        // mid SAGE: xk2 = relu(sage(xk)) [N, 256]
        launch_agg(xk, 256, esrc, edst, agg, stream);
        launch_gemm(1, agg, xk, invdeg, Wl_m, bl_m, Wr_m, xk2, 256, 256, 256, stream);

        // output SAGE: hk = sigmoid(sage(xk2)) [N, 64], stored over xk buffer
        float* hk = xk;
        launch_agg(xk2, 256, esrc, edst, agg, stream);
        launch_gemm(2, agg, xk2, invdeg, Wl_o, bl_o, Wr_o, hk, 64, 256, 64, stream);

        // heads share one aggregation of hk
        launch_agg(hk, 64, esrc, edst, agg, stream);
        for (int h = 0; h < 3; ++h) {
            const int C = C_h[h];
            float* o = xk2;   // [N, 16] scratch for head logits
            launch_gemm(2, agg, hk, invdeg,
                        Wl_h[h] + (size_t)k * 64 * C,
                        bl_h[h] + (size_t)k * C,
                        Wr_h[h] + (size_t)k * 64 * C,
                        o, 16, 64, C, stream);
            ce_kernel<<<(N_NODES + 255) / 256, 256, 0, stream>>>(
                o, 16, y + (size_t)k * N_NODES * 3 + h, C, N_NODES, (float*)d_out);
        }
    }
}